// IDWT_2D_19164144075606
// MI455X (gfx1250) — compile-verified
//
#include <hip/hip_runtime.h>
#include <stdint.h>

// IDWT-2D (db2) on gfx1250: banded-window f32 WMMA GEMMs + async-to-LDS staging.
// Stage 1: column synthesis  L = LL*ml1^T + LH*mh1^T  (Hb from HL/HH)
// Stage 2: row synthesis     out = ml0^T*L + mh0^T*Hb
// Each stage: 16x16 = (16xK)*(Kx16) GEMM, K=12 band window -> 3 chained
// v_wmma_f32_16x16x4_f32 per matrix pair (18 WMMAs per output tile).

typedef __attribute__((ext_vector_type(2))) float v2f;
typedef __attribute__((ext_vector_type(8))) float v8f;

__device__ __forceinline__ v8f wmma4(v2f a, v2f b, v8f c) {
  // (neg_a, A, neg_b, B, c_mod, C, reuse_a, reuse_b)
  return __builtin_amdgcn_wmma_f32_16x16x4_f32(false, a, false, b, (short)0, c,
                                               false, false);
}

// Branch-free guarded load: clamp index, unconditional load, cndmask result.
__device__ __forceinline__ float sel_ld(const float* p, int idx, int lim) {
  int cl = idx < 0 ? 0 : (idx >= lim - 1 ? lim - 1 : idx);
  float v = p[cl];
  return (idx == cl) ? v : 0.0f;
}

// m[(k)*pitch + col] when 0 <= k < kmax, else 0  (branch-free)
__device__ __forceinline__ float sel_row(const float* m, int k, int col,
                                         int pitch, int kmax) {
  bool ok = (unsigned)k < (unsigned)kmax;
  int kc = ok ? k : 0;
  float v = m[(size_t)kc * pitch + col];
  return ok ? v : 0.0f;
}

__global__ __launch_bounds__(256) void idwt2d_db2_wmma(
    const float* __restrict__ LL, const float* __restrict__ LH,
    const float* __restrict__ HL, const float* __restrict__ HH,
    const float* __restrict__ ml0, const float* __restrict__ ml1,
    const float* __restrict__ mh0, const float* __restrict__ mh1,
    float* __restrict__ out, int nimg) {
  constexpr int H2 = 256, W2 = 256, H = 512, W = 512;
  constexpr int JW = 76;            // padded j-window stride (floats); 76%64=12
  constexpr int SUBSZ = 16 * JW;    // 1216 floats per subband window

  // 4 subband windows (19456 B) + 8 waves * two 16x16 f32 tiles (16384 B)
  __shared__ float smem[4 * SUBSZ + 8 * 512];

  const int tid  = threadIdx.x;
  const int wave = tid >> 5;
  const int lane = tid & 31;
  const int r    = lane & 15;   // M for A frags / N for B,C,D frags
  const int kh   = lane >> 4;   // half-wave: K pair {0,1} vs {2,3}

  int bid = blockIdx.x;
  const int wstrip = bid & 3;   bid >>= 2;   // 4 strips of 128 cols
  const int htile  = bid & 31;                // 32 tiles of 16 rows
  const int img    = bid >> 5;

  const int h0 = htile << 4;
  const int wb = wstrip << 7;          // block column base
  const int w0 = wb + (wave << 4);     // this wave's 16-col tile

  const int kb  = (h0 >> 1) - 1;       // row-stage K window: k in [kb, kb+11]
  const int jal = (wb >> 1) - 4;       // 16B-aligned j base of staged window
  const int jbw = (w0 >> 1) - 1;       // this wave's col-stage j window base

  const size_t simg = (size_t)img * (H2 * W2);

  // ---------------- Stage 0: stage subband windows into LDS ----------------
  const bool interior =
      (kb >= 0) && (kb + 15 < H2) && (jal >= 0) && (jal + JW - 1 < W2);

  if (interior) {
    // Fully in-bounds: coalesced 128-bit async copies global -> LDS.
    for (int i = tid; i < 4 * 16 * (JW / 4); i += 256) {  // 1216 f4 chunks
      const int c4   = i % (JW / 4);
      const int rb   = i / (JW / 4);
      const int row  = rb & 15;
      const int band = rb >> 4;
      const float* bp = band == 0 ? LL : band == 1 ? LH : band == 2 ? HL : HH;
      const float* src = bp + simg + (size_t)(kb + row) * W2 + (jal + 4 * c4);
      uint32_t lva =
          (uint32_t)(uintptr_t)&smem[band * SUBSZ + row * JW + 4 * c4];
      asm volatile("global_load_async_to_lds_b128 %0, %1, off"
                   :: "v"(lva), "v"(src) : "memory");
    }
    asm volatile("s_wait_asynccnt 0" ::: "memory");
  } else {
    // Boundary blocks: branch-free clamp+select scalar staging with zero fill.
    for (int i = tid; i < 4 * SUBSZ; i += 256) {
      const int col  = i % JW;
      const int rb   = i / JW;
      const int row  = rb & 15;
      const int band = rb >> 4;
      const float* bp = band == 0 ? LL : band == 1 ? LH : band == 2 ? HL : HH;
      const int k = kb + row, j = jal + col;
      const bool ok = ((unsigned)k < (unsigned)H2) && ((unsigned)j < (unsigned)W2);
      const int kc = ok ? k : 0, jc = ok ? j : 0;
      const float v = bp[simg + (size_t)kc * W2 + jc];
      smem[band * SUBSZ + row * JW + col] = ok ? v : 0.0f;
    }
  }
  __syncthreads();

  // ---------------- Stage 1: column synthesis (K window = 12) --------------
  const float* pml1 = ml1 + (size_t)(w0 + r) * W2;  // ml1 is (W, W/2) row-major
  const float* pmh1 = mh1 + (size_t)(w0 + r) * W2;
  const int jbase = (wave << 3) + 3;                // jbw - jal

  v8f cL = {0.f, 0.f, 0.f, 0.f, 0.f, 0.f, 0.f, 0.f};
  v8f cH = {0.f, 0.f, 0.f, 0.f, 0.f, 0.f, 0.f, 0.f};

#pragma unroll
  for (int s = 0; s < 3; ++s) {
    const int o  = r * JW + jbase + 4 * s + 2 * kh;  // A frag: LDS offset
    const int j0 = jbw + 4 * s + 2 * kh;             // B frag: global j

    v2f aLL = {smem[0 * SUBSZ + o], smem[0 * SUBSZ + o + 1]};
    v2f aLH = {smem[1 * SUBSZ + o], smem[1 * SUBSZ + o + 1]};
    v2f aHL = {smem[2 * SUBSZ + o], smem[2 * SUBSZ + o + 1]};
    v2f aHH = {smem[3 * SUBSZ + o], smem[3 * SUBSZ + o + 1]};

    v2f bL = {sel_ld(pml1, j0, W2), sel_ld(pml1, j0 + 1, W2)};
    v2f bH = {sel_ld(pmh1, j0, W2), sel_ld(pmh1, j0 + 1, W2)};

    cL = wmma4(aLL, bL, cL);   // LL * ml1^T
    cL = wmma4(aLH, bH, cL);   // + LH * mh1^T
    cH = wmma4(aHL, bL, cH);   // HL * ml1^T
    cH = wmma4(aHH, bH, cH);   // + HH * mh1^T
  }

  // ---------------- C/D layout -> B layout via LDS --------------------------
  float* myL = smem + 4 * SUBSZ + wave * 512;
  float* myH = myL + 256;
#pragma unroll
  for (int i = 0; i < 8; ++i) {
    myL[(i + 8 * kh) * 16 + r] = cL[i];
    myH[(i + 8 * kh) * 16 + r] = cH[i];
  }
  __syncthreads();

  // ---------------- Stage 2: row synthesis (K window = 12) ------------------
  v8f acc = {0.f, 0.f, 0.f, 0.f, 0.f, 0.f, 0.f, 0.f};
#pragma unroll
  for (int s = 0; s < 3; ++s) {
    const int rk = 4 * s + 2 * kh;   // K index within window
    const int kk = kb + rk;          // global k row of ml0/mh0 (H/2 x H)

    v2f a2l = {sel_row(ml0, kk, h0 + r, W, H2),
               sel_row(ml0, kk + 1, h0 + r, W, H2)};
    v2f a2h = {sel_row(mh0, kk, h0 + r, W, H2),
               sel_row(mh0, kk + 1, h0 + r, W, H2)};

    v2f b2l = {myL[rk * 16 + r], myL[(rk + 1) * 16 + r]};
    v2f b2h = {myH[rk * 16 + r], myH[(rk + 1) * 16 + r]};

    acc = wmma4(a2l, b2l, acc);   // ml0^T * L
    acc = wmma4(a2h, b2h, acc);   // + mh0^T * Hb
  }

  // ---------------- Store 16x16 output tile (D layout) ----------------------
  float* po = out + (size_t)img * (H * W) + (size_t)h0 * W + w0;
#pragma unroll
  for (int i = 0; i < 8; ++i) po[(size_t)(i + 8 * kh) * W + r] = acc[i];
}

extern "C" void kernel_launch(void* const* d_in, const int* in_sizes, int n_in,
                              void* d_out, int out_size, void* d_ws,
                              size_t ws_size, hipStream_t stream) {
  const float* LL  = (const float*)d_in[0];
  const float* LH  = (const float*)d_in[1];
  const float* HL  = (const float*)d_in[2];
  const float* HH  = (const float*)d_in[3];
  const float* ml0 = (const float*)d_in[4];
  const float* ml1 = (const float*)d_in[5];
  const float* mh0 = (const float*)d_in[6];
  const float* mh1 = (const float*)d_in[7];
  float* out = (float*)d_out;

  const int nimg = in_sizes[0] / (256 * 256);   // B*C = 512
  const int grid = nimg * 32 * 4;               // h-tiles * w-strips
  idwt2d_db2_wmma<<<grid, 256, 0, stream>>>(LL, LH, HL, HH, ml0, ml1, mh0, mh1,
                                            out, nimg);
}